// AdditiveAttention_14955076124747
// MI455X (gfx1250) — compile-verified
//
#include <hip/hip_runtime.h>
#include <hip/hip_bf16.h>
#include <math.h>

#define B_  16
#define Q_  64
#define K_  1024
#define QS_ 256
#define H_  128
#define VS_ 256
#define NEG_ (-1e6f)

typedef float v2f __attribute__((ext_vector_type(2)));
typedef float v8f __attribute__((ext_vector_type(8)));

// Prefer the gfx1250 hardware transcendentals (single TRANS-rate ops).
#if defined(__has_builtin)
#  if __has_builtin(__builtin_amdgcn_tanhf)
#    define TANHF(x) __builtin_amdgcn_tanhf(x)
#  else
#    define TANHF(x) tanhf(x)
#  endif
#  if __has_builtin(__builtin_amdgcn_exp2f)
#    define EXP2F(x) __builtin_amdgcn_exp2f(x)
#  else
#    define EXP2F(x) exp2f(x)
#  endif
#else
#  define TANHF(x) tanhf(x)
#  define EXP2F(x) exp2f(x)
#endif

#define LOG2E 1.4426950408889634f

// ---------------------------------------------------------------------------
// Row-major fp32 GEMM using V_WMMA_F32_16X16X4_F32.
// One wave computes one 16x16 tile of C = A(MxKD) * W(KDxN).
// N, KD compile-time -> strided B loads become immediate-offset vmem ops.
// blockDim.x = 128 (4 waves). blockIdx.y = batch (float strides sA/sW/sC).
// Requires N%16==0, KD%4==0, total tiles % 4 == 0.
// ---------------------------------------------------------------------------
template<int N, int KD>
__global__ __launch_bounds__(128) void gemm_wmma_f32(
    const float* __restrict__ A, long sA,
    const float* __restrict__ W, long sW,
    float* __restrict__ C, long sC)
{
    A += (size_t)blockIdx.y * sA;
    W += (size_t)blockIdx.y * sW;
    C += (size_t)blockIdx.y * sC;

    const int wave = threadIdx.x >> 5;
    const int lane = threadIdx.x & 31;
    const int tile = blockIdx.x * 4 + wave;
    constexpr int tilesN = N >> 4;
    const int tm = tile / tilesN;          // power-of-2 -> shift
    const int tn = tile - tm * tilesN;
    const int r    = lane & 15;            // row/col within tile
    const int half = lane >> 4;            // 0: K+{0,1}, 1: K+{2,3}

    const float* arow = A + (size_t)(tm * 16 + r) * KD + 2 * half;
    const float* wcol = W + (size_t)(2 * half) * N + tn * 16 + r;

    v8f c = {};
    #pragma unroll 8
    for (int kk = 0; kk < KD; kk += 4) {
        const v2f a = *(const v2f*)(arow + kk);          // global_load_b64
        v2f b;
        b.x = wcol[(size_t)kk * N];                      // imm-offset b32
        b.y = wcol[(size_t)(kk + 1) * N];
        c = __builtin_amdgcn_wmma_f32_16x16x4_f32(
                /*neg_a=*/false, a, /*neg_b=*/false, b,
                /*c_mod=*/(short)0, c, /*reuse_a=*/false, /*reuse_b=*/false);
    }

    // C/D layout: VGPR v -> M = v + 8*half, N = r
    float* crow = C + (size_t)(tm * 16 + 8 * half) * N + tn * 16 + r;
    #pragma unroll
    for (int v = 0; v < 8; ++v)
        crow[(size_t)v * N] = c[v];
}

// ---------------------------------------------------------------------------
// Fused: scores[b,q,k] = sum_h Ws[h] * tanh(qproj[b,q,h] + kproj[b,k,h]),
// front-mask (k < valid_len[b] -> -1e6), softmax over k. One block per (b,q).
// 256 threads = 8 waves; each wave owns one k per iteration, 32 lanes x
// float4 cover H=128 contiguously (coalesced 512B per wave per k).
// ---------------------------------------------------------------------------
__global__ __launch_bounds__(256) void scores_softmax(
    const float* __restrict__ qproj,   // (B*Q, H)
    const float* __restrict__ kproj,   // (B*K, H)
    const float* __restrict__ Ws,      // (H)
    const int*   __restrict__ valid_len,
    float* __restrict__ attn)          // (B*Q, K)
{
    __shared__ float s_scores[K_];
    __shared__ float s_red[8];

    const int b = blockIdx.x >> 6;     // / Q_
    const int q = blockIdx.x & 63;     // % Q_
    const int t = threadIdx.x;
    const int wave = t >> 5;
    const int lane = t & 31;
    const int h0 = lane * 4;

    const float4 qv = *(const float4*)(qproj + ((size_t)b * Q_ + q) * H_ + h0);
    const float4 wv = *(const float4*)(Ws + h0);
    const int vlen = valid_len[b];
    const float* kbase = kproj + (size_t)b * K_ * H_;

    for (int k = wave; k < K_; k += 8) {
        const float4 kv = *(const float4*)(kbase + (size_t)k * H_ + h0);
        float p = wv.x * TANHF(qv.x + kv.x)
                + wv.y * TANHF(qv.y + kv.y)
                + wv.z * TANHF(qv.z + kv.z)
                + wv.w * TANHF(qv.w + kv.w);
        #pragma unroll
        for (int off = 16; off >= 1; off >>= 1)
            p += __shfl_xor(p, off, 32);
        if (lane == 0)
            s_scores[k] = (k < vlen) ? NEG_ : p;
    }
    __syncthreads();

    // Softmax over 1024 scores: 4 elements per thread.
    const float v0 = s_scores[t];
    const float v1 = s_scores[t + 256];
    const float v2 = s_scores[t + 512];
    const float v3 = s_scores[t + 768];

    float m = fmaxf(fmaxf(v0, v1), fmaxf(v2, v3));
    #pragma unroll
    for (int off = 16; off >= 1; off >>= 1)
        m = fmaxf(m, __shfl_xor(m, off, 32));
    if (lane == 0) s_red[wave] = m;
    __syncthreads();
    float gmax = s_red[0];
    #pragma unroll
    for (int w = 1; w < 8; ++w) gmax = fmaxf(gmax, s_red[w]);
    __syncthreads();  // s_red reused below

    const float e0 = EXP2F((v0 - gmax) * LOG2E);
    const float e1 = EXP2F((v1 - gmax) * LOG2E);
    const float e2 = EXP2F((v2 - gmax) * LOG2E);
    const float e3 = EXP2F((v3 - gmax) * LOG2E);
    float s = e0 + e1 + e2 + e3;
    #pragma unroll
    for (int off = 16; off >= 1; off >>= 1)
        s += __shfl_xor(s, off, 32);
    if (lane == 0) s_red[wave] = s;
    __syncthreads();
    float tot = s_red[0];
    #pragma unroll
    for (int w = 1; w < 8; ++w) tot += s_red[w];
    const float inv = 1.0f / tot;

    float* orow = attn + ((size_t)b * Q_ + q) * K_;
    orow[t]       = e0 * inv;
    orow[t + 256] = e1 * inv;
    orow[t + 512] = e2 * inv;
    orow[t + 768] = e3 * inv;
}

// ---------------------------------------------------------------------------
extern "C" void kernel_launch(void* const* d_in, const int* in_sizes, int n_in,
                              void* d_out, int out_size, void* d_ws, size_t ws_size,
                              hipStream_t stream) {
    const float* queries   = (const float*)d_in[0];
    const float* keys      = (const float*)d_in[1];
    const float* values    = (const float*)d_in[2];
    const int*   valid_len = (const int*)  d_in[3];
    const float* Wq        = (const float*)d_in[4];
    const float* Wk        = (const float*)d_in[5];
    const float* Ws        = (const float*)d_in[6];
    float* out = (float*)d_out;

    float* qproj = (float*)d_ws;                               // B*Q*H
    float* kproj = qproj + (size_t)B_ * Q_ * H_;               // B*K*H
    float* attn  = kproj + (size_t)B_ * K_ * H_;               // B*Q*K

    // q = queries @ Wq : (1024 x 256)(256 x 128). 512 tiles -> 128 blocks.
    gemm_wmma_f32<H_, QS_><<<dim3(128, 1), 128, 0, stream>>>(
        queries, 0, Wq, 0, qproj, 0);

    // k = keys @ Wk : (16384 x 256)(256 x 128). 8192 tiles -> 2048 blocks.
    gemm_wmma_f32<H_, QS_><<<dim3(2048, 1), 128, 0, stream>>>(
        keys, 0, Wk, 0, kproj, 0);

    // Fused tanh-score + mask + softmax. One block per (b,q).
    scores_softmax<<<B_ * Q_, 256, 0, stream>>>(
        qproj, kproj, Ws, valid_len, attn);

    // out[b] = attn[b] @ values[b] : (64 x 1024)(1024 x 256), batched over B.
    gemm_wmma_f32<VS_, K_><<<dim3(16, B_), 128, 0, stream>>>(
        attn,   (long)Q_ * K_,
        values, (long)K_ * VS_,
        out,    (long)Q_ * VS_);
}